// CTFLoss_35407710388469
// MI455X (gfx1250) — compile-verified
//
#include <hip/hip_runtime.h>

#define B_   64
#define T_   1024
#define C_   256
#define N_   128
#define S_   257      // 2N+1 extended states
#define SP_  264      // padded state stride
#define NEG_ (-1e30f)

typedef __attribute__((ext_vector_type(2))) float v2f;
typedef __attribute__((ext_vector_type(8))) float v8f;

__device__ __forceinline__ float lse3(float a, float b, float c) {
    float m = fmaxf(a, fmaxf(b, c));
    return m + logf(expf(a - m) + expf(b - m) + expf(c - m));
}

// Build extended labels (blank-interleaved), can_skip flags, return label length.
// Uses shared arrays ext[SP_], csk[SP_], cnt. Called by all threads (block 288).
__device__ __forceinline__ int build_ext(const int* __restrict__ Tg, int b, int tid,
                                         int* ext, int* csk, int* cnt) {
    if (tid == 0) *cnt = 0;
    __syncthreads();
    if (tid < N_) {
        int v = Tg[b * N_ + tid];
        ext[2 * tid + 1] = (v >= 0) ? v : 0;
        if (v >= 0) atomicAdd(cnt, 1);
    }
    if (tid <= N_) ext[2 * tid] = 0;          // blanks at even positions (incl. s=2N)
    if (tid >= S_ && tid < SP_) ext[tid] = 0; // pad
    __syncthreads();
    if (tid < SP_) {
        int e = ext[tid];
        csk[tid] = (tid >= 2) && (tid < S_) && (e != 0) && (e != ext[tid - 2]);
    }
    __syncthreads();
    return *cnt;
}

// ---------------- K1: log-softmax stats per (b,t) row: m and log-sum-exp --------------
__global__ __launch_bounds__(256) void k_lse(const float* __restrict__ O,
                                             float2* __restrict__ mlse) {
    int lane = threadIdx.x & 31, wid = threadIdx.x >> 5;
    long row = (long)blockIdx.x * 8 + wid;            // one wave per row of C_=256
    const float4* p = (const float4*)(O + row * C_);
    float4 x0 = p[lane];
    float4 x1 = p[lane + 32];
    float m = fmaxf(fmaxf(fmaxf(x0.x, x0.y), fmaxf(x0.z, x0.w)),
                    fmaxf(fmaxf(x1.x, x1.y), fmaxf(x1.z, x1.w)));
#pragma unroll
    for (int i = 16; i > 0; i >>= 1) m = fmaxf(m, __shfl_xor(m, i));
    float s = expf(x0.x - m) + expf(x0.y - m) + expf(x0.z - m) + expf(x0.w - m)
            + expf(x1.x - m) + expf(x1.y - m) + expf(x1.z - m) + expf(x1.w - m);
#pragma unroll
    for (int i = 16; i > 0; i >>= 1) s += __shfl_xor(s, i);
    if (lane == 0) mlse[row] = make_float2(m, logf(s));
}

// ---------------- K2: CTC beta (backward) recursion, store beta[T][SP_], logP ---------
__global__ __launch_bounds__(288) void k_beta(const float* __restrict__ O,
                                              const int* __restrict__ Tg,
                                              const float2* __restrict__ mlse,
                                              float* __restrict__ beta,
                                              float* __restrict__ logPv) {
    __shared__ int   ext[SP_];
    __shared__ int   csk[SP_];
    __shared__ float bb[2][SP_];
    __shared__ int   cnt;
    int b = blockIdx.x, tid = threadIdx.x;
    int L = build_ext(Tg, b, tid, ext, csk, &cnt);
    bool valid = tid < S_;
    int s = valid ? tid : (S_ - 1);
    int e_s = ext[s];
    int fin0 = 2 * L;
    int fin1 = (2 * L - 1 > 0) ? (2 * L - 1) : 0;

    // t = T-1 init
    {
        int t = T_ - 1;
        float2 ml = mlse[b * T_ + t];
        float emit = O[((size_t)b * T_ + t) * C_ + e_s] - ml.x - ml.y;
        float cur = (valid && (s == fin0 || s == fin1)) ? emit : NEG_;
        if (valid) {
            bb[0][s] = cur;
            beta[((size_t)b * T_ + t) * SP_ + s] = cur;
        }
        __syncthreads();
    }
    int cb = 0;
    for (int t = T_ - 2; t >= 0; --t) {
        float2 ml = mlse[b * T_ + t];
        float emit = O[((size_t)b * T_ + t) * C_ + e_s] - ml.x - ml.y;
        float* prev = bb[cb];
        float a  = prev[s];
        float bv = (s + 1 < S_) ? prev[s + 1] : NEG_;
        float cv = (s + 2 < S_ && csk[s + 2]) ? prev[s + 2] : NEG_;
        float cur = emit + lse3(a, bv, cv);
        if (valid) {
            bb[cb ^ 1][s] = cur;
            beta[((size_t)b * T_ + t) * SP_ + s] = cur;
        }
        cb ^= 1;
        __syncthreads();
    }
    if (tid == 0) {
        float l1 = bb[cb][0], l2 = bb[cb][1];
        float m = fmaxf(l1, l2);
        logPv[b] = m + logf(expf(l1 - m) + expf(l2 - m));
    }
}

// ---------------- K3: alpha forward + class scatter + focal loss (WMMA f32 accum) -----
__global__ __launch_bounds__(288) void k_alpha_loss(const float* __restrict__ O,
                                                    const int* __restrict__ Tg,
                                                    const float2* __restrict__ mlse,
                                                    const float* __restrict__ beta,
                                                    const float* __restrict__ logPv,
                                                    float* __restrict__ out) {
    __shared__ int   ext[SP_];
    __shared__ int   csk[SP_];
    __shared__ float ab[2][SP_];
    __shared__ float W[SP_];     // W[0..255]: per-class scatter; W[S_]: sum of w
    __shared__ float wred[8];
    __shared__ int   cnt;
    int b = blockIdx.x, tid = threadIdx.x;
    int lane = tid & 31;
    (void)build_ext(Tg, b, tid, ext, csk, &cnt);
    bool valid = tid < S_;
    int s = valid ? tid : (S_ - 1);
    int e_s = ext[s];
    float logP = logPv[b];

    v8f acc = {0.f, 0.f, 0.f, 0.f, 0.f, 0.f, 0.f, 0.f};
    v2f ones; ones.x = 1.0f; ones.y = 1.0f;

    for (int t = 0; t < T_; ++t) {
        const float* orow = O + ((size_t)b * T_ + t) * C_;
        float2 ml = mlse[b * T_ + t];
        float emit = orow[e_s] - ml.x - ml.y;
        if (tid < SP_) W[tid] = 0.0f;
        float cur;
        if (t == 0) {
            cur = (s < 2) ? emit : NEG_;
        } else {
            float* prev = ab[(t & 1) ^ 1];
            float a  = prev[s];
            float bv = (s >= 1) ? prev[s - 1] : NEG_;
            float cv = csk[s] ? prev[s - 2] : NEG_;
            cur = emit + lse3(a, bv, cv);
        }
        float bt = beta[((size_t)b * T_ + t) * SP_ + s];
        if (t + 1 < T_)  // prefetch next beta row (global_prefetch_b8)
            __builtin_prefetch(&beta[((size_t)b * T_ + t + 1) * SP_ + s], 0, 0);
        float w = valid ? expf(cur + bt - emit - logP) : 0.0f;
        __syncthreads();                           // W zeroed; prev-buffer reads done
        if (valid) ab[t & 1][s] = cur;
        // wave-level pre-reduction: total w (for gsum) and blank (even-s) w
        float wAll = w;
        float wBlank = ((s & 1) == 0) ? w : 0.0f;  // invalid lanes carry w==0
#pragma unroll
        for (int i = 16; i > 0; i >>= 1) {
            wAll   += __shfl_xor(wAll, i);
            wBlank += __shfl_xor(wBlank, i);
        }
        if (lane == 0) {
            atomicAdd(&W[S_], wAll);               // ds_add_f32
            if (wBlank != 0.0f) atomicAdd(&W[0], wBlank);
        }
        if (valid && (s & 1)) atomicAdd(&W[e_s], w);
        __syncthreads();                           // scatter complete
        if (tid < C_) {                            // waves 0..7 fully active -> EXEC all 1s
            float lp = orow[tid] - ml.x - ml.y;
            float p  = expf(lp);
            float Wc = W[tid];
            float sw = W[S_];
            float gl = p * sw - Wc;                // grad_logits
            float ce = p - gl;                     // == stop_grad target
            float cw = fmaxf(fabsf(gl), 1e-8f);    // focal weight, gamma=1
            v2f a2; a2.x = cw * ce * lp; a2.y = 0.0f;
            // fp32 matrix accumulate: D = A x ones + C  (row-sum engine)
            acc = __builtin_amdgcn_wmma_f32_16x16x4_f32(
                false, a2, false, ones, (short)0, acc, false, false);
        }
        __syncthreads();                           // W consumed before next zeroing
    }
    if (tid < C_) {
        float lsum = acc[0] + acc[1] + acc[2] + acc[3]
                   + acc[4] + acc[5] + acc[6] + acc[7];
#pragma unroll
        for (int i = 16; i > 0; i >>= 1) lsum += __shfl_xor(lsum, i);
        if (lane == 0) wred[tid >> 5] = lsum;
    }
    __syncthreads();
    if (tid == 0) {
        float tot = 0.0f;
#pragma unroll
        for (int i = 0; i < 8; ++i) tot += wred[i];
        // WMMA replicated each row-sum across 16 columns -> divide by 16; loss = -sum
        atomicAdd(out, -tot * 0.0625f);
    }
}

extern "C" void kernel_launch(void* const* d_in, const int* in_sizes, int n_in,
                              void* d_out, int out_size, void* d_ws, size_t ws_size,
                              hipStream_t stream) {
    const float* O  = (const float*)d_in[0];   // [B,T,C] float32
    const int*   Tg = (const int*)d_in[1];     // [B,N]  int32
    float* out = (float*)d_out;

    char* ws = (char*)d_ws;
    size_t o_mlse = 0;
    size_t o_logP = o_mlse + (size_t)B_ * T_ * sizeof(float2);   // 512 KB
    size_t o_beta = o_logP + 256;                                // 64 floats, padded
    float2* mlse = (float2*)(ws + o_mlse);
    float*  logP = (float*)(ws + o_logP);
    float*  beta = (float*)(ws + o_beta);                        // B*T*SP_ floats (~66 MB)

    hipMemsetAsync(d_out, 0, sizeof(float), stream);
    k_lse<<<(B_ * T_) / 8, 256, 0, stream>>>(O, mlse);
    k_beta<<<B_, 288, 0, stream>>>(O, Tg, mlse, beta, logP);
    k_alpha_loss<<<B_, 288, 0, stream>>>(O, Tg, mlse, beta, logP, out);
}